// EdgeGATLayer_3229815406927
// MI455X (gfx1250) — compile-verified
//
#include <hip/hip_runtime.h>
#include <math.h>

// Problem constants (match reference)
#define B_  8
#define OP_ 400
#define MA_ 200
#define F_  64

typedef float v2f __attribute__((ext_vector_type(2)));
typedef float v8f __attribute__((ext_vector_type(8)));

// ---------------------------------------------------------------------------
// K1: Y(R x 64) = X(R x 64) @ W(64 x 64), fp32 WMMA 16x16x4.
// One wave per 16x16 output tile; blockDim=128 (4 waves = 4 N-tiles),
// gridDim = R/16 (M tiles). Grid sized exactly -> EXEC all ones for WMMA.
// ---------------------------------------------------------------------------
__global__ void gemm64_wmma(const float* __restrict__ X,
                            const float* __restrict__ W,
                            float* __restrict__ Y) {
  const int lane = threadIdx.x & 31;
  const int tn   = threadIdx.x >> 5;        // N tile 0..3
  const int tm   = blockIdx.x;              // M tile
  const int l15  = lane & 15;
  const int kh   = (lane >> 4) << 1;        // 0 or 2 (K half select)
  const int row  = tm * 16 + l15;
  const int col  = tn * 16 + l15;
  v8f acc = {0.f,0.f,0.f,0.f,0.f,0.f,0.f,0.f};
#pragma unroll
  for (int kb = 0; kb < F_; kb += 4) {
    v2f a, b;
    a.x = X[(size_t)row * F_ + kb + kh];
    a.y = X[(size_t)row * F_ + kb + kh + 1];
    b.x = W[(size_t)(kb + kh) * F_ + col];
    b.y = W[(size_t)(kb + kh + 1) * F_ + col];
    acc = __builtin_amdgcn_wmma_f32_16x16x4_f32(false, a, false, b,
                                                (short)0, acc, false, false);
  }
  const int rbase = tm * 16 + ((lane >> 4) << 3);   // row = rbase + v
#pragma unroll
  for (int v = 0; v < 8; ++v)
    Y[(size_t)(rbase + v) * F_ + col] = acc[v];
}

// ---------------------------------------------------------------------------
// K2: small dot products:
//   xa[b,n] = h_op . av[0:64], ya[b,m] = h . av[64:128], w_e = W_edge @ av[128:192]
// ---------------------------------------------------------------------------
__global__ void prep_dots(const float* __restrict__ h_op,
                          const float* __restrict__ h,
                          const float* __restrict__ W_edge,
                          const float* __restrict__ av,
                          float* __restrict__ xa,
                          float* __restrict__ ya,
                          float* __restrict__ w_e) {
  const int idx = blockIdx.x * blockDim.x + threadIdx.x;
  if (idx < B_ * OP_) {
    float s = 0.f;
#pragma unroll
    for (int o = 0; o < F_; ++o) s += h_op[(size_t)idx * F_ + o] * av[o];
    xa[idx] = s;
  } else if (idx < B_ * OP_ + B_ * MA_) {
    const int r = idx - B_ * OP_;
    float s = 0.f;
#pragma unroll
    for (int o = 0; o < F_; ++o) s += h[(size_t)r * F_ + o] * av[F_ + o];
    ya[r] = s;
  } else if (idx < B_ * OP_ + B_ * MA_ + F_) {
    const int i = idx - B_ * OP_ - B_ * MA_;
    float s = 0.f;
#pragma unroll
    for (int o = 0; o < F_; ++o) s += W_edge[(size_t)i * F_ + o] * av[2 * F_ + o];
    w_e[i] = s;
  }
}

// ---------------------------------------------------------------------------
// K3 (pass 1 over z, adjacency-gated): e[b,m,n] =
//   leaky_relu(xa[b,n] + ya[b,m] + z[b,n,m,:].w_e)  where adj>0, else sentinel.
// One wave per (b,n,m): 256B contiguous float2 read per lane + shfl reduce.
// ---------------------------------------------------------------------------
__global__ void pass1_e(const float* __restrict__ z,
                        const int* __restrict__ adj,
                        const float* __restrict__ xa,
                        const float* __restrict__ ya,
                        const float* __restrict__ we,
                        float* __restrict__ e) {
  const int gw   = (int)((blockIdx.x * blockDim.x + threadIdx.x) >> 5); // (b*OP+n)*MA+m
  const int lane = threadIdx.x & 31;
  const int m  = gw % MA_;
  const int bn = gw / MA_;       // b*OP_+n
  const int b  = bn / OP_;
  const int n  = bn % OP_;
  float val;
  if (adj[gw] > 0) {             // wave-uniform: skip z read for masked edges
    const float2 zv = ((const float2*)(z + (size_t)gw * F_))[lane];
    const float2 wv = ((const float2*)we)[lane];
    float s = zv.x * wv.x + zv.y * wv.y;
#pragma unroll
    for (int off = 16; off; off >>= 1) s += __shfl_xor(s, off, 32);
    val = xa[bn] + ya[b * MA_ + m] + s;
    val = val > 0.f ? val : 0.01f * val;   // leaky_relu(0.01)
  } else {
    val = -3.0e38f;                        // never read by softmax (masked)
  }
  if (lane == 0) e[((size_t)(b * MA_ + m)) * OP_ + n] = val;
}

// ---------------------------------------------------------------------------
// K4: masked softmax over n (in place on e), one wave per (b,m) row.
// Empty rows: att=0 + flag (output is re-masked later, matching reference).
// ---------------------------------------------------------------------------
__global__ void softmax_rows(float* __restrict__ e,
                             const int* __restrict__ adj,
                             int* __restrict__ empty) {
  const int row  = (int)((blockIdx.x * blockDim.x + threadIdx.x) >> 5); // b*MA+m
  const int lane = threadIdx.x & 31;
  const int b = row / MA_, m = row % MA_;
  float* er = e + (size_t)row * OP_;
  const int* ar = adj + (size_t)b * OP_ * MA_ + m;   // stride MA_ over n
  float ev[13];
  unsigned vm = 0;
#pragma unroll
  for (int k = 0; k < 13; ++k) {
    const int n = lane + k * 32;
    if (n < OP_) {
      ev[k] = er[n];
      if (ar[(size_t)n * MA_] > 0) vm |= 1u << k;
    }
  }
  float mx = -INFINITY;
#pragma unroll
  for (int k = 0; k < 13; ++k)
    if (vm & (1u << k)) mx = fmaxf(mx, ev[k]);
#pragma unroll
  for (int off = 16; off; off >>= 1) mx = fmaxf(mx, __shfl_xor(mx, off, 32));
  if (!(mx > -INFINITY)) {                  // wave-uniform: empty row
#pragma unroll
    for (int k = 0; k < 13; ++k) { const int n = lane + k * 32; if (n < OP_) er[n] = 0.f; }
    if (lane == 0) empty[row] = 1;
    return;
  }
  float ex[13];
  float s = 0.f;
#pragma unroll
  for (int k = 0; k < 13; ++k) {
    const float v = (vm & (1u << k)) ? __expf(ev[k] - mx) : 0.f;
    ex[k] = v; s += v;
  }
#pragma unroll
  for (int off = 16; off; off >>= 1) s += __shfl_xor(s, off, 32);
  const float inv = 1.f / s;
#pragma unroll
  for (int k = 0; k < 13; ++k) { const int n = lane + k * 32; if (n < OP_) er[n] = ex[k] * inv; }
  if (lane == 0) empty[row] = 0;
}

// ---------------------------------------------------------------------------
// K5 (pass 2 over z, att-gated; z is L2-resident after pass 1):
//   zbar[b,m,i] = sum_n att[b,m,n] * z[b,n,m,i]
//   hbar[b,m,o] = sum_n att[b,m,n] * h_op[b,n,o]
// One block per (b,m); thread = (n-phase 0..3, i 0..63); LDS tree for phases.
// ---------------------------------------------------------------------------
__global__ void pass2_agg(const float* __restrict__ z,
                          const float* __restrict__ hop,
                          const float* __restrict__ att,
                          float* __restrict__ zbar,
                          float* __restrict__ hbar) {
  const int row = blockIdx.x;               // b*MA+m
  const int b = row / MA_, m = row % MA_;
  const int t  = threadIdx.x;
  const int i  = t & 63;
  const int ns = t >> 6;                    // 0..3
  float az = 0.f, ah = 0.f;
  const float* ar = att + (size_t)row * OP_;
  for (int n = ns; n < OP_; n += 4) {
    if (n + 4 < OP_)  // strided stream: prefetch next z row (global_prefetch_b8)
      __builtin_prefetch(&z[(((size_t)(b * OP_ + n + 4)) * MA_ + m) * F_ + i], 0, 1);
    const float w = ar[n];
    if (w != 0.f) {                         // uniform across the 64 i-threads
      az += w * z[(((size_t)(b * OP_ + n)) * MA_ + m) * F_ + i];
      ah += w * hop[(size_t)(b * OP_ + n) * F_ + i];
    }
  }
  __shared__ float sz[4][F_];
  __shared__ float sh[4][F_];
  sz[ns][i] = az; sh[ns][i] = ah;
  __syncthreads();
  if (t < F_) {
    zbar[(size_t)row * F_ + t] = sz[0][t] + sz[1][t] + sz[2][t] + sz[3][t];
    hbar[(size_t)row * F_ + t] = sh[0][t] + sh[1][t] + sh[2][t] + sh[3][t];
  }
}

// ---------------------------------------------------------------------------
// K6: out = elu( mask(hbar + zbar @ W_edge) + h ), fp32 WMMA + fused epilogue.
// ---------------------------------------------------------------------------
__global__ void gemm64_wmma_final(const float* __restrict__ Zb,
                                  const float* __restrict__ W,
                                  const float* __restrict__ Hb,
                                  const float* __restrict__ Hm,
                                  const int* __restrict__ empty,
                                  float* __restrict__ out) {
  const int lane = threadIdx.x & 31;
  const int tn   = threadIdx.x >> 5;
  const int tm   = blockIdx.x;
  const int l15  = lane & 15;
  const int kh   = (lane >> 4) << 1;
  const int row  = tm * 16 + l15;
  const int col  = tn * 16 + l15;
  v8f acc = {0.f,0.f,0.f,0.f,0.f,0.f,0.f,0.f};
#pragma unroll
  for (int kb = 0; kb < F_; kb += 4) {
    v2f a, b;
    a.x = Zb[(size_t)row * F_ + kb + kh];
    a.y = Zb[(size_t)row * F_ + kb + kh + 1];
    b.x = W[(size_t)(kb + kh) * F_ + col];
    b.y = W[(size_t)(kb + kh + 1) * F_ + col];
    acc = __builtin_amdgcn_wmma_f32_16x16x4_f32(false, a, false, b,
                                                (short)0, acc, false, false);
  }
  const int rbase = tm * 16 + ((lane >> 4) << 3);
#pragma unroll
  for (int v = 0; v < 8; ++v) {
    const int r = rbase + v;
    float c = acc[v] + Hb[(size_t)r * F_ + col];
    if (empty[r]) c = 0.f;                   // row_empty mask before +h
    c += Hm[(size_t)r * F_ + col];
    out[(size_t)r * F_ + col] = c > 0.f ? c : (__expf(c) - 1.f);  // ELU(alpha=1)
  }
}

// ---------------------------------------------------------------------------
extern "C" void kernel_launch(void* const* d_in, const int* in_sizes, int n_in,
                              void* d_out, int out_size, void* d_ws, size_t ws_size,
                              hipStream_t stream) {
  const float* x      = (const float*)d_in[0];  // (8,400,64)
  const float* y      = (const float*)d_in[1];  // (8,200,64)
  const float* z      = (const float*)d_in[2];  // (8,400,200,64)
  const int*   adj    = (const int*)  d_in[3];  // (8,400,200)
  const float* W_op   = (const float*)d_in[4];
  const float* W_ma   = (const float*)d_in[5];
  const float* W_edge = (const float*)d_in[6];
  const float* av     = (const float*)d_in[7];  // (192,)
  float* out = (float*)d_out;                   // (8,200,64)

  // Workspace layout (floats); total ~4.6 MB
  float* ws    = (float*)d_ws;
  float* h_op  = ws;                       // 8*400*64 = 204800
  float* h     = h_op + 204800;            // 8*200*64 = 102400
  float* xa    = h    + 102400;            // 3200
  float* ya    = xa   + 3200;              // 1600
  float* w_e   = ya   + 1600;              // 64
  float* e     = w_e  + 64;                // 8*200*400 = 640000 (e, then att in place)
  float* zbar  = e    + 640000;            // 102400
  float* hbar  = zbar + 102400;            // 102400
  int*   empty = (int*)(hbar + 102400);    // 1600

  gemm64_wmma<<<200, 128, 0, stream>>>(x, W_op, h_op);          // 3200x64x64
  gemm64_wmma<<<100, 128, 0, stream>>>(y, W_ma, h);             // 1600x64x64
  prep_dots  <<<19, 256, 0, stream>>>(h_op, h, W_edge, av, xa, ya, w_e);
  pass1_e    <<<80000, 256, 0, stream>>>(z, adj, xa, ya, w_e, e);
  softmax_rows<<<200, 256, 0, stream>>>(e, adj, empty);
  pass2_agg  <<<1600, 256, 0, stream>>>(z, h_op, e, zbar, hbar);
  gemm64_wmma_final<<<100, 128, 0, stream>>>(zbar, W_edge, hbar, h, empty, out);
}